// GNNStack_14714557956710
// MI455X (gfx1250) — compile-verified
//
#include <hip/hip_runtime.h>
#include <hip/hip_bf16.h>
#include <math.h>

// ---------------- constants from reference ----------------
#define NN      40000      // nodes
#define E0E     640000     // original edges
#define EE      680000     // edges + self loops
#define DIN     128
#define HIDC    64
#define HEADS   4
#define HCC     256        // HEADS*HID
#define NGR     256
#define DOUTC   10
#define NEGS    0.2f
#define LNEPS   1e-5f
#define GEMM_KC 64         // K-chunk staged in LDS

typedef float v2f __attribute__((ext_vector_type(2)));
typedef float v8f __attribute__((ext_vector_type(8)));

// ordered-uint encoding for float atomic max (handles negatives)
__device__ __forceinline__ unsigned fenc(float f) {
    unsigned u = __float_as_uint(f);
    return (u & 0x80000000u) ? ~u : (u | 0x80000000u);
}
__device__ __forceinline__ float fdec(unsigned u) {
    return (u & 0x80000000u) ? __uint_as_float(u & 0x7FFFFFFFu)
                             : __uint_as_float(~u);
}

// ---------------- self-loop attr = mean of incoming edge attrs ----------------
__global__ void selfloop_sum(const int* __restrict__ ei, const float* __restrict__ eattr,
                             float* __restrict__ loopAttr, float* __restrict__ cnt) {
    int e = blockIdx.x * blockDim.x + threadIdx.x;
    if (e >= E0E) return;
    int dst = ei[E0E + e];
    atomicAdd(&cnt[dst], 1.0f);
    atomicAdd(&loopAttr[dst * 3 + 0], eattr[e * 3 + 0]);
    atomicAdd(&loopAttr[dst * 3 + 1], eattr[e * 3 + 1]);
    atomicAdd(&loopAttr[dst * 3 + 2], eattr[e * 3 + 2]);
}
__global__ void selfloop_div(float* __restrict__ loopAttr, const float* __restrict__ cnt) {
    int i = blockIdx.x * blockDim.x + threadIdx.x;
    if (i >= NN) return;
    float c = fmaxf(cnt[i], 1.0f);
    loopAttr[i * 3 + 0] /= c;
    loopAttr[i * 3 + 1] /= c;
    loopAttr[i * 3 + 2] /= c;
}

// ---------------- WMMA fp32 GEMM: out[r,0..255] = X[r,:K] @ W[K,256] + bias ----------------
// One 512-thread block (16 wave32s) per 16-row output panel: each wave owns one
// 16x16 col-tile. The 16xGEMM_KC X chunk is staged once in LDS (coalesced) and
// shared by all 16 waves; next chunk is prefetched (global_prefetch_b8).
// A fragment: lane L holds (M=L%16, K=2*(L/16)+v); B: (K=2*(L/16)+v, N=L%16);
// C/D: (M=r+8*(L/16), N=L%16)  [CDNA5 ISA layouts]
__global__ __launch_bounds__(512)
void wmma_gemm_bias(const float* __restrict__ X, const float* __restrict__ W,
                    const float* __restrict__ bias, float* __restrict__ out,
                    int K) {
    __shared__ float sX[16][GEMM_KC + 1];   // +1 pad: kill LDS bank conflicts
    const int waveId = threadIdx.x >> 5;    // 0..15 = col-tile
    const int lane   = threadIdx.x & 31;
    const int row0   = blockIdx.x * 16;     // 2500 row panels, exact
    const int m    = lane & 15;
    const int kh   = lane >> 4;             // K-half selector (0/1)
    const int colg = waveId * 16 + m;
    v8f acc = {};
    for (int k0 = 0; k0 < K; k0 += GEMM_KC) {
        __syncthreads();                    // previous chunk fully consumed
        for (int idx = threadIdx.x; idx < 16 * GEMM_KC; idx += 512) {
            int r = idx >> 6, c = idx & (GEMM_KC - 1);
            sX[r][c] = X[(size_t)(row0 + r) * K + k0 + c];
        }
        if (k0 + GEMM_KC < K) {             // prefetch next chunk into caches
            for (int idx = threadIdx.x; idx < 16 * GEMM_KC; idx += 512) {
                int r = idx >> 6, c = idx & (GEMM_KC - 1);
                __builtin_prefetch(&X[(size_t)(row0 + r) * K + k0 + GEMM_KC + c], 0, 0);
            }
        }
        __syncthreads();
#pragma unroll
        for (int kc = 0; kc < GEMM_KC; kc += 4) {
            int kb = kc + 2 * kh;
            v2f a, b;
            a.x = sX[m][kb + 0];
            a.y = sX[m][kb + 1];
            b.x = W[(size_t)(k0 + kb + 0) * HCC + colg];
            b.y = W[(size_t)(k0 + kb + 1) * HCC + colg];
            acc = __builtin_amdgcn_wmma_f32_16x16x4_f32(
                false, a, false, b, (short)0, acc, false, false);
        }
    }
    float bv = bias[colg];
#pragma unroll
    for (int r = 0; r < 8; r++) {
        out[(size_t)(row0 + r + 8 * kh) * HCC + colg] = acc[r] + bv;
    }
}

// ---------------- edge pass 1: alpha[e,h] + segment max over dst ----------------
__global__ void edge_alpha(const int* __restrict__ ei, const float* __restrict__ eattr,
                           const float* __restrict__ loopAttr,
                           const float* __restrict__ xl, const float* __restrict__ xr,
                           const float* __restrict__ We, const float* __restrict__ att,
                           float* __restrict__ alpha, unsigned* __restrict__ amaxU) {
    __shared__ float sWe[3 * HCC];
    __shared__ float sAtt[HCC];
    for (int i = threadIdx.x; i < 3 * HCC; i += blockDim.x) sWe[i] = We[i];
    for (int i = threadIdx.x; i < HCC; i += blockDim.x) sAtt[i] = att[i];
    __syncthreads();
    int tid = blockIdx.x * blockDim.x + threadIdx.x;
    if (tid >= EE * HEADS) return;
    int e = tid >> 2, h = tid & 3;
    int src, dst; float a0, a1, a2;
    if (e < E0E) {
        src = ei[e]; dst = ei[E0E + e];
        a0 = eattr[e * 3 + 0]; a1 = eattr[e * 3 + 1]; a2 = eattr[e * 3 + 2];
    } else {
        src = dst = e - E0E;
        a0 = loopAttr[src * 3 + 0]; a1 = loopAttr[src * 3 + 1]; a2 = loopAttr[src * 3 + 2];
    }
    const float* xls = xl + (size_t)src * HCC + h * HIDC;
    const float* xrd = xr + (size_t)dst * HCC + h * HIDC;
    float s = 0.0f;
#pragma unroll 4
    for (int c = 0; c < HIDC; c++) {
        int cc = h * HIDC + c;
        float m = xls[c] + xrd[c] + a0 * sWe[cc] + a1 * sWe[HCC + cc] + a2 * sWe[2 * HCC + cc];
        m = (m > 0.0f) ? m : NEGS * m;
        s += sAtt[cc] * m;
    }
    alpha[tid] = s;
    atomicMax(&amaxU[dst * HEADS + h], fenc(s));
}

__global__ void amax_decode(const unsigned* __restrict__ amaxU, float* __restrict__ amaxF) {
    int i = blockIdx.x * blockDim.x + threadIdx.x;
    if (i >= NN * HEADS) return;
    unsigned u = amaxU[i];
    amaxF[i] = (u == 0u) ? 0.0f : fdec(u);
}

// ---------------- edge pass 2: denominator ----------------
__global__ void edge_expsum(const int* __restrict__ ei, const float* __restrict__ alpha,
                            const float* __restrict__ amaxF, float* __restrict__ den) {
    int tid = blockIdx.x * blockDim.x + threadIdx.x;
    if (tid >= EE * HEADS) return;
    int e = tid >> 2, h = tid & 3;
    int dst = (e < E0E) ? ei[E0E + e] : (e - E0E);
    float ex = expf(alpha[tid] - amaxF[dst * HEADS + h]);
    atomicAdd(&den[dst * HEADS + h], ex);
}

// ---------------- edge pass 3: normalize alpha in place ----------------
__global__ void edge_norm(const int* __restrict__ ei, float* __restrict__ alpha,
                          const float* __restrict__ amaxF, const float* __restrict__ den) {
    int tid = blockIdx.x * blockDim.x + threadIdx.x;
    if (tid >= EE * HEADS) return;
    int e = tid >> 2, h = tid & 3;
    int dst = (e < E0E) ? ei[E0E + e] : (e - E0E);
    alpha[tid] = expf(alpha[tid] - amaxF[dst * HEADS + h]) / den[dst * HEADS + h];
}

// ---------------- edge pass 4: weighted scatter (one block per edge) ----------------
__global__ void edge_scatter(const int* __restrict__ ei, const float* __restrict__ alpha,
                             const float* __restrict__ xl, float* __restrict__ conv) {
    int e = blockIdx.x;
    int cc = threadIdx.x;            // 0..255
    int h = cc >> 6;
    int src, dst;
    if (e < E0E) { src = ei[e]; dst = ei[E0E + e]; }
    else         { src = dst = e - E0E; }
    float w = alpha[e * HEADS + h];
    atomicAdd(&conv[(size_t)dst * HCC + cc], w * xl[(size_t)src * HCC + cc]);
}

// ---------------- layer1 epilogue: +bias, relu, layernorm (wave32 per node) ----------------
__global__ void post_ln(const float* __restrict__ conv, const float* __restrict__ bias,
                        const float* __restrict__ g, const float* __restrict__ b,
                        float* __restrict__ hout) {
    int node = blockIdx.x * 8 + (threadIdx.x >> 5);
    int lane = threadIdx.x & 31;
    if (node >= NN) return;
    float vals[8];
    float sum = 0.0f, sumsq = 0.0f;
#pragma unroll
    for (int j = 0; j < 8; j++) {
        int c = j * 32 + lane;
        float v = conv[(size_t)node * HCC + c] + bias[c];
        v = fmaxf(v, 0.0f);
        vals[j] = v;
        sum += v; sumsq += v * v;
    }
#pragma unroll
    for (int off = 16; off > 0; off >>= 1) {
        sum   += __shfl_xor(sum, off, 32);
        sumsq += __shfl_xor(sumsq, off, 32);
    }
    float mu  = sum * (1.0f / HCC);
    float var = sumsq * (1.0f / HCC) - mu * mu;
    float rs  = rsqrtf(var + LNEPS);
#pragma unroll
    for (int j = 0; j < 8; j++) {
        int c = j * 32 + lane;
        hout[(size_t)node * HCC + c] = (vals[j] - mu) * rs * g[c] + b[c];
    }
}

// ---------------- layer2 epilogue: emb = conv + bias (written to d_out) ----------------
__global__ void post_emb(const float* __restrict__ conv, const float* __restrict__ bias,
                         float* __restrict__ emb) {
    int i = blockIdx.x * blockDim.x + threadIdx.x;
    if (i >= NN * HCC) return;
    emb[i] = conv[i] + bias[i & (HCC - 1)];
}

// ---------------- global max pool per graph over relu(emb) ----------------
__global__ void pool_max(const float* __restrict__ emb, const int* __restrict__ batch,
                         unsigned* __restrict__ gpool) {
    int i = blockIdx.x * blockDim.x + threadIdx.x;
    if (i >= NN * HCC) return;
    int node = i >> 8, c = i & (HCC - 1);
    float v = fmaxf(emb[i], 0.0f);      // >=0: raw bits order == float order
    atomicMax(&gpool[batch[node] * HCC + c], __float_as_uint(v));
}

// ---------------- head MLP + log_softmax (one block per graph) ----------------
__global__ void mlp_head(const unsigned* __restrict__ gpool,
                         const float* __restrict__ W1, const float* __restrict__ b1,
                         const float* __restrict__ W2, const float* __restrict__ b2,
                         const float* __restrict__ W3, const float* __restrict__ b3,
                         float* __restrict__ outLogits) {
    __shared__ float gs[HCC], h1[HIDC], h2[HIDC], lg[DOUTC];
    int gi = blockIdx.x, t = threadIdx.x;   // 64 threads
    for (int i = t; i < HCC; i += 64) gs[i] = __uint_as_float(gpool[gi * HCC + i]);
    __syncthreads();
    float s = b1[t];
    for (int c = 0; c < HCC; c++) s += gs[c] * W1[c * HIDC + t];
    h1[t] = fmaxf(s, 0.0f);
    __syncthreads();
    s = b2[t];
    for (int c = 0; c < HIDC; c++) s += h1[c] * W2[c * HIDC + t];
    h2[t] = fmaxf(s, 0.0f);
    __syncthreads();
    if (t < DOUTC) {
        s = b3[t];
        for (int c = 0; c < HIDC; c++) s += h2[c] * W3[c * DOUTC + t];
        lg[t] = s;
    }
    __syncthreads();
    if (t == 0) {
        float mx = lg[0];
        for (int k = 1; k < DOUTC; k++) mx = fmaxf(mx, lg[k]);
        float se = 0.0f;
        for (int k = 0; k < DOUTC; k++) se += expf(lg[k] - mx);
        float lse = logf(se);
        for (int k = 0; k < DOUTC; k++)
            outLogits[gi * DOUTC + k] = lg[k] - mx - lse;
    }
}

// ---------------- host side ----------------
static void run_attention(hipStream_t stream, const int* ei, const float* eattr,
                          const float* loopAttr, const float* xl, const float* xr,
                          const float* We, const float* att,
                          float* alpha, unsigned* amaxU, float* amaxF, float* den,
                          float* conv) {
    hipMemsetAsync(amaxU, 0, (size_t)NN * HEADS * sizeof(unsigned), stream);
    hipMemsetAsync(den, 0, (size_t)NN * HEADS * sizeof(float), stream);
    hipMemsetAsync(conv, 0, (size_t)NN * HCC * sizeof(float), stream);
    int eh = EE * HEADS;
    edge_alpha<<<(eh + 255) / 256, 256, 0, stream>>>(ei, eattr, loopAttr, xl, xr, We, att, alpha, amaxU);
    amax_decode<<<(NN * HEADS + 255) / 256, 256, 0, stream>>>(amaxU, amaxF);
    edge_expsum<<<(eh + 255) / 256, 256, 0, stream>>>(ei, alpha, amaxF, den);
    edge_norm<<<(eh + 255) / 256, 256, 0, stream>>>(ei, alpha, amaxF, den);
    edge_scatter<<<EE, HCC, 0, stream>>>(ei, alpha, xl, conv);
}

extern "C" void kernel_launch(void* const* d_in, const int* in_sizes, int n_in,
                              void* d_out, int out_size, void* d_ws, size_t ws_size,
                              hipStream_t stream) {
    const float* x      = (const float*)d_in[0];
    const int*   ei     = (const int*)d_in[1];
    const float* eattr  = (const float*)d_in[2];
    const int*   batch  = (const int*)d_in[3];
    const float* c1Wl   = (const float*)d_in[4];
    const float* c1bl   = (const float*)d_in[5];
    const float* c1Wr   = (const float*)d_in[6];
    const float* c1br   = (const float*)d_in[7];
    const float* c1We   = (const float*)d_in[8];
    const float* c1att  = (const float*)d_in[9];
    const float* c1bias = (const float*)d_in[10];
    const float* c2Wl   = (const float*)d_in[11];
    const float* c2bl   = (const float*)d_in[12];
    const float* c2Wr   = (const float*)d_in[13];
    const float* c2br   = (const float*)d_in[14];
    const float* c2We   = (const float*)d_in[15];
    const float* c2att  = (const float*)d_in[16];
    const float* c2bias = (const float*)d_in[17];
    const float* ln_g   = (const float*)d_in[18];
    const float* ln_b   = (const float*)d_in[19];
    const float* W1     = (const float*)d_in[20];
    const float* b1     = (const float*)d_in[21];
    const float* W2     = (const float*)d_in[22];
    const float* b2     = (const float*)d_in[23];
    const float* W3     = (const float*)d_in[24];
    const float* b3     = (const float*)d_in[25];

    float* ws = (float*)d_ws;
    float*    loopAttr = ws;                         // NN*3
    float*    cnt      = loopAttr + (size_t)NN * 3;  // NN
    float*    xl       = cnt + NN;                   // NN*256
    float*    xr       = xl + (size_t)NN * HCC;      // NN*256
    float*    conv     = xr + (size_t)NN * HCC;      // NN*256
    float*    hbuf     = conv + (size_t)NN * HCC;    // NN*256
    float*    alpha    = hbuf + (size_t)NN * HCC;    // EE*4
    unsigned* amaxU    = (unsigned*)(alpha + (size_t)EE * HEADS); // NN*4
    float*    amaxF    = (float*)(amaxU + (size_t)NN * HEADS);    // NN*4
    float*    den      = amaxF + (size_t)NN * HEADS; // NN*4
    unsigned* gpool    = (unsigned*)(den + (size_t)NN * HEADS);   // 256*256

    float* outEmb    = (float*)d_out;
    float* outLogits = outEmb + (size_t)NN * HCC;

    // self-loop attributes (fill_value='mean')
    hipMemsetAsync(loopAttr, 0, (size_t)NN * 3 * sizeof(float), stream);
    hipMemsetAsync(cnt, 0, (size_t)NN * sizeof(float), stream);
    selfloop_sum<<<(E0E + 255) / 256, 256, 0, stream>>>(ei, eattr, loopAttr, cnt);
    selfloop_div<<<(NN + 255) / 256, 256, 0, stream>>>(loopAttr, cnt);

    // ---- layer 1 ----
    int panels = NN / 16;                  // 2500 row panels, exact
    wmma_gemm_bias<<<panels, 512, 0, stream>>>(x, c1Wl, c1bl, xl, DIN);
    wmma_gemm_bias<<<panels, 512, 0, stream>>>(x, c1Wr, c1br, xr, DIN);
    run_attention(stream, ei, eattr, loopAttr, xl, xr, c1We, c1att,
                  alpha, amaxU, amaxF, den, conv);
    post_ln<<<NN / 8, 256, 0, stream>>>(conv, c1bias, ln_g, ln_b, hbuf);

    // ---- layer 2 ----
    wmma_gemm_bias<<<panels, 512, 0, stream>>>(hbuf, c2Wl, c2bl, xl, HCC);
    wmma_gemm_bias<<<panels, 512, 0, stream>>>(hbuf, c2Wr, c2br, xr, HCC);
    run_attention(stream, ei, eattr, loopAttr, xl, xr, c2We, c2att,
                  alpha, amaxU, amaxF, den, conv);
    post_emb<<<(NN * HCC + 255) / 256, 256, 0, stream>>>(conv, c2bias, outEmb);

    // ---- pooling + head MLP ----
    hipMemsetAsync(gpool, 0, (size_t)NGR * HCC * sizeof(unsigned), stream);
    pool_max<<<(NN * HCC + 255) / 256, 256, 0, stream>>>(outEmb, batch, gpool);
    mlp_head<<<NGR, HIDC, 0, stream>>>(gpool, W1, b1, W2, b2, W3, b3, outLogits);
}